// SGNN_Attention_64716567216295
// MI455X (gfx1250) — compile-verified
//
#include <hip/hip_runtime.h>
#include <hip/hip_bf16.h>

#define F 128
#define H 128
#define C 10
#define LH 3
#define NEG_SLOPE 0.2f

typedef float v2f __attribute__((ext_vector_type(2)));
typedef float v8f __attribute__((ext_vector_type(8)));
typedef unsigned int u32x4 __attribute__((ext_vector_type(4)));
typedef int i32x8 __attribute__((ext_vector_type(8)));
typedef int i32x4 __attribute__((ext_vector_type(4)));

// ---------- helpers ----------
__device__ __forceinline__ unsigned f32_to_ordered_u32(float f) {
    unsigned u = __float_as_uint(f);
    return (u & 0x80000000u) ? ~u : (u | 0x80000000u);
}
__device__ __forceinline__ float ordered_u32_to_f32(unsigned u) {
    unsigned b = (u & 0x80000000u) ? (u ^ 0x80000000u) : ~u;
    return __uint_as_float(b);
}
__device__ __forceinline__ void atom_add_f32(float* p, float v) {
    unsafeAtomicAdd(p, v);   // lowers to global_atomic_add_f32 on gfx1250
}

// ---------- init per-node softmax state ----------
__global__ __launch_bounds__(256) void init_kernel(unsigned* __restrict__ menc,
                                                   float* __restrict__ ssum, int n) {
    int i = blockIdx.x * blockDim.x + threadIdx.x;
    if (i < n) {
        menc[i] = 0x007FFFFFu;   // ordered encoding of -inf
        ssum[i] = 0.0f;
    }
}

// ---------- xs = x@a_src, xd = x@a_dst (wave per node) ----------
__global__ __launch_bounds__(256) void node_dots_kernel(const float* __restrict__ x,
                                                        const float* __restrict__ asrc,
                                                        const float* __restrict__ adst,
                                                        float* __restrict__ xs,
                                                        float* __restrict__ xd, int n) {
    int wid  = (blockIdx.x * blockDim.x + threadIdx.x) >> 5;
    int lane = threadIdx.x & 31;
    if (wid >= n) return;
    float4 v = ((const float4*)(x + (size_t)wid * F))[lane];
    float4 s = ((const float4*)asrc)[lane];
    float4 d = ((const float4*)adst)[lane];
    float ps = v.x * s.x + v.y * s.y + v.z * s.z + v.w * s.w;
    float pd = v.x * d.x + v.y * d.y + v.z * d.z + v.w * d.w;
    for (int off = 16; off; off >>= 1) {
        ps += __shfl_xor(ps, off, 32);
        pd += __shfl_xor(pd, off, 32);
    }
    if (lane == 0) { xs[wid] = ps; xd[wid] = pd; }
}

// ---------- edge logits + segment max ----------
__global__ __launch_bounds__(256) void edge_max_kernel(const int* __restrict__ src,
                                                       const int* __restrict__ dst,
                                                       const float* __restrict__ xs,
                                                       const float* __restrict__ xd,
                                                       float* __restrict__ eraw,
                                                       unsigned* __restrict__ menc, int E) {
    int e = blockIdx.x * blockDim.x + threadIdx.x;
    if (e >= E) return;
    float v = xs[src[e]] + xd[dst[e]];
    float lr = (v > 0.0f) ? v : NEG_SLOPE * v;
    eraw[e] = lr;
    atomicMax(&menc[dst[e]], f32_to_ordered_u32(lr));
}

// ---------- exp(e - max) + segment sum ----------
__global__ __launch_bounds__(256) void edge_exp_kernel(const int* __restrict__ dst,
                                                       const float* __restrict__ eraw,
                                                       const unsigned* __restrict__ menc,
                                                       float* __restrict__ att,
                                                       float* __restrict__ ssum, int E) {
    int e = blockIdx.x * blockDim.x + threadIdx.x;
    if (e >= E) return;
    int d   = dst[e];
    float m = ordered_u32_to_f32(menc[d]);
    float ex = __expf(eraw[e] - m);
    att[e] = ex;
    atom_add_f32(&ssum[d], ex);
}

// ---------- normalize ----------
__global__ __launch_bounds__(256) void edge_norm_kernel(const int* __restrict__ dst,
                                                        float* __restrict__ att,
                                                        const float* __restrict__ ssum, int E) {
    int e = blockIdx.x * blockDim.x + threadIdx.x;
    if (e >= E) return;
    att[e] = att[e] / (ssum[dst[e]] + 1e-16f);
}

// ---------- symmetrize all 2*(LH+1) weight matrices: S = 0.5*(W + W^T) ----------
__global__ __launch_bounds__(256) void symm_kernel(const float* __restrict__ Wup,
                                                   const float* __restrict__ Wdn,
                                                   float* __restrict__ S) {
    int idx = blockIdx.x * blockDim.x + threadIdx.x;      // over (LH+1)*H*H
    if (idx >= (LH + 1) * H * H) return;
    int i = idx >> 14;
    int r = (idx >> 7) & 127;
    int c = idx & 127;
    const float* Wu = Wup + (size_t)i * H * H;
    const float* Wd = Wdn + (size_t)i * H * H;
    S[(size_t)(2 * i) * H * H + r * H + c]     = 0.5f * (Wu[r * H + c] + Wu[c * H + r]);
    S[(size_t)(2 * i + 1) * H * H + r * H + c] = 0.5f * (Wd[r * H + c] + Wd[c * H + r]);
}

// ---------- C[N,128] = A[N,128] @ B[128,128], fp32 WMMA 16x16x4 ----------
// B staged into LDS by the Tensor Data Mover (tensor_load_to_lds, TENSORcnt).
// block = 128 threads (4 waves); each wave owns column tiles {w, w+4}
__global__ __launch_bounds__(128) void gemm_n128_kernel(const float* __restrict__ A,
                                                        const float* __restrict__ B,
                                                        float* __restrict__ Cm, int nrows) {
    __shared__ float Bs[H * H];

    if (threadIdx.x == 0) {
        unsigned lds = (unsigned)(uintptr_t)&Bs[0];       // low 32 bits of generic ptr = LDS offset
        unsigned long long ga = (unsigned long long)(uintptr_t)B;
        // D# group0: count=1 | lds_addr | global_addr[56:0] | type=2
        u32x4 g0;
        g0.x = 1u;
        g0.y = lds;
        g0.z = (unsigned)(ga & 0xffffffffu);
        g0.w = (unsigned)((ga >> 32) & 0x01ffffffu) | (2u << 30);
        // D# group1: data_size=4B | tensor_dim0=128 | tensor_dim1=128 |
        //            tile_dim0=128 | tile_dim1=128 | tensor_dim0_stride=128
        i32x8 g1;
        g1[0] = 0x00020000;          // data_size=2 (4 bytes), no iterate/pad/multicast
        g1[1] = (int)(128u << 16);   // tensor_dim0[15:0] at bits 63:48
        g1[2] = (int)(128u << 16);   // tensor_dim1[15:0] at bits 95:80
        g1[3] = (int)(128u << 16);   // tile_dim0 at bits 127:112
        g1[4] = 128;                 // tile_dim1 at bits 143:128
        g1[5] = 128;                 // tensor_dim0_stride low 32 at bits 191:160
        g1[6] = 0;
        g1[7] = 0;
        i32x4 gz4 = {0, 0, 0, 0};
        i32x8 gz8 = {0, 0, 0, 0, 0, 0, 0, 0};
        __builtin_amdgcn_tensor_load_to_lds(g0, g1, gz4, gz4, gz8, 0);
        __builtin_amdgcn_s_wait_tensorcnt(0);
    }
    __syncthreads();

    int tile = blockIdx.x;                 // 16-row tile
    int wave = threadIdx.x >> 5;
    int lane = threadIdx.x & 31;
    int half = lane >> 4;
    int l    = lane & 15;
    int row  = tile * 16 + l;
    int rowc = (row < nrows) ? row : (nrows - 1);   // clamp loads, keep EXEC full for WMMA
    const float* Arow = A + (size_t)rowc * H;
    int c0 = wave * 16;                    // second tile at c0+64

    v8f acc0 = {0.f, 0.f, 0.f, 0.f, 0.f, 0.f, 0.f, 0.f};
    v8f acc1 = {0.f, 0.f, 0.f, 0.f, 0.f, 0.f, 0.f, 0.f};

    for (int k = 0; k < H; k += 4) {
        int kb = k + 2 * half;
        v2f a;
        a.x = Arow[kb];
        a.y = Arow[kb + 1];
        const float* B0 = &Bs[kb * H];
        v2f b0, b1;
        b0.x = B0[c0 + l];
        b0.y = B0[H + c0 + l];
        b1.x = B0[c0 + 64 + l];
        b1.y = B0[H + c0 + 64 + l];
        acc0 = __builtin_amdgcn_wmma_f32_16x16x4_f32(false, a, false, b0, (short)0, acc0, false, false);
        acc1 = __builtin_amdgcn_wmma_f32_16x16x4_f32(false, a, false, b1, (short)0, acc1, false, false);
    }

    for (int r = 0; r < 8; ++r) {
        int m = r + 8 * half;
        int orow = tile * 16 + m;
        if (orow < nrows) {
            Cm[(size_t)orow * H + c0 + l]      = acc0[r];
            Cm[(size_t)orow * H + c0 + 64 + l] = acc1[r];
        }
    }
}

// ---------- edge scatter: out[dst] += att[e] * u[src]  (wave per edge) ----------
__global__ __launch_bounds__(256) void scatter_kernel(const float* __restrict__ u,
                                                      const int* __restrict__ src,
                                                      const int* __restrict__ dst,
                                                      const float* __restrict__ att,
                                                      float* __restrict__ out, int E) {
    int wid  = (blockIdx.x * blockDim.x + threadIdx.x) >> 5;
    int lane = threadIdx.x & 31;
    if (wid >= E) return;
    int s = src[wid];
    int d = dst[wid];
    float a = att[wid];
    float4 v = ((const float4*)(u + (size_t)s * H))[lane];
    float* o = out + (size_t)d * H + lane * 4;
    atom_add_f32(o + 0, a * v.x);
    atom_add_f32(o + 1, a * v.y);
    atom_add_f32(o + 2, a * v.z);
    atom_add_f32(o + 3, a * v.w);
}

// ---------- fused activations: p += au*tanh(q); q += ad*tanh(p_new) ----------
__global__ __launch_bounds__(256) void act_kernel(float* __restrict__ p, float* __restrict__ q,
                                                  const float* __restrict__ au,
                                                  const float* __restrict__ ad, size_t total) {
    size_t i = (size_t)blockIdx.x * blockDim.x + threadIdx.x;
    if (i >= total) return;
    int h = (int)(i & (H - 1));
    float qv = q[i];
    float pv = p[i] + au[h] * tanhf(qv);
    p[i] = pv;
    q[i] = qv + ad[h] * tanhf(pv);
}

// ---------- out = q @ W_out + b_out (wave per node) ----------
__global__ __launch_bounds__(256) void out_kernel(const float* __restrict__ q,
                                                  const float* __restrict__ Wout,
                                                  const float* __restrict__ bout,
                                                  float* __restrict__ out, int n) {
    int wid  = (blockIdx.x * blockDim.x + threadIdx.x) >> 5;
    int lane = threadIdx.x & 31;
    if (wid >= n) return;
    float4 v = ((const float4*)(q + (size_t)wid * H))[lane];
    int k = lane * 4;
    for (int c = 0; c < C; ++c) {
        float acc = v.x * Wout[k * C + c] + v.y * Wout[(k + 1) * C + c] +
                    v.z * Wout[(k + 2) * C + c] + v.w * Wout[(k + 3) * C + c];
        for (int off = 16; off; off >>= 1) acc += __shfl_xor(acc, off, 32);
        if (lane == 0) out[(size_t)wid * C + c] = acc + bout[c];
    }
}

extern "C" void kernel_launch(void* const* d_in, const int* in_sizes, int n_in,
                              void* d_out, int out_size, void* d_ws, size_t ws_size,
                              hipStream_t stream) {
    const float* x      = (const float*)d_in[0];
    const int*   ei     = (const int*)d_in[1];
    const float* a_src  = (const float*)d_in[2];
    const float* a_dst  = (const float*)d_in[3];
    const float* Wp     = (const float*)d_in[4];
    const float* Wq     = (const float*)d_in[5];
    const float* W_up   = (const float*)d_in[6];
    const float* W_dn   = (const float*)d_in[7];
    const float* act_up = (const float*)d_in[8];
    const float* act_dn = (const float*)d_in[9];
    const float* W_out  = (const float*)d_in[10];
    const float* b_out  = (const float*)d_in[11];
    float* out = (float*)d_out;

    const int N = in_sizes[0] / F;
    const int E = in_sizes[1] / 2;
    const int* src = ei;
    const int* dst = ei + E;

    // workspace carve-up (floats)
    float* ws = (float*)d_ws;
    float*    p    = ws;
    float*    q    = p + (size_t)N * H;
    float*    u    = q + (size_t)N * H;
    float*    xs   = u + (size_t)N * H;
    float*    xd   = xs + N;
    unsigned* menc = (unsigned*)(xd + N);
    float*    ssum = (float*)(menc + N);
    float*    eraw = ssum + N;
    float*    att  = eraw + E;
    float*    S    = att + E;   // 2*(LH+1) matrices of H*H

    const int eb = (E + 255) / 256;
    const int nb = (N + 255) / 256;
    const int wn = (N + 7) / 8;     // wave-per-node blocks (8 waves/block)
    const int we = (E + 7) / 8;     // wave-per-edge blocks
    const int gt = (N + 15) / 16;   // gemm row tiles

    // 1) attention
    init_kernel<<<nb, 256, 0, stream>>>(menc, ssum, N);
    node_dots_kernel<<<wn, 256, 0, stream>>>(x, a_src, a_dst, xs, xd, N);
    edge_max_kernel<<<eb, 256, 0, stream>>>(src, dst, xs, xd, eraw, menc, E);
    edge_exp_kernel<<<eb, 256, 0, stream>>>(dst, eraw, menc, att, ssum, E);
    edge_norm_kernel<<<eb, 256, 0, stream>>>(dst, att, ssum, E);

    // 2) symmetrized weights
    symm_kernel<<<((LH + 1) * H * H + 255) / 256, 256, 0, stream>>>(W_up, W_dn, S);

    // 3) p = x@Wp, q = x@Wq
    gemm_n128_kernel<<<gt, 128, 0, stream>>>(x, Wp, p, N);
    gemm_n128_kernel<<<gt, 128, 0, stream>>>(x, Wq, q, N);

    // 4) message passing
    for (int i = 0; i <= LH; ++i) {
        const float* Sup = S + (size_t)(2 * i) * H * H;
        const float* Sdn = S + (size_t)(2 * i + 1) * H * H;
        // p += segsum(att * (q@Sup)[src])
        gemm_n128_kernel<<<gt, 128, 0, stream>>>(q, Sup, u, N);
        scatter_kernel<<<we, 256, 0, stream>>>(u, src, dst, att, p, E);
        // q += segsum(att * (p@Sdn)[src])
        gemm_n128_kernel<<<gt, 128, 0, stream>>>(p, Sdn, u, N);
        scatter_kernel<<<we, 256, 0, stream>>>(u, src, dst, att, q, E);
        if (i < LH) {
            act_kernel<<<(int)(((size_t)N * H + 255) / 256), 256, 0, stream>>>(
                p, q, act_up + (size_t)i * H, act_dn + (size_t)i * H, (size_t)N * H);
        }
    }

    // 5) classifier
    out_kernel<<<wn, 256, 0, stream>>>(q, W_out, b_out, out, N);
}